// ACCLoss_25366076850443
// MI455X (gfx1250) — compile-verified
//
#include <hip/hip_runtime.h>
#include <math.h>

typedef __attribute__((ext_vector_type(2))) float v2f;
typedef __attribute__((ext_vector_type(8))) float v8f;

#define B_DIM 4096
#define E_DIM 64
#define D_DIM 256

__global__ void acc_zero_kernel(int* __restrict__ out) {
    if (threadIdx.x < 24) out[threadIdx.x] = 0;
}

__launch_bounds__(128)
__global__ void acc_main_kernel(const float* __restrict__ predicted,
                                const float* __restrict__ ev,
                                const int*   __restrict__ em,
                                int*         __restrict__ out) {
    __shared__ float lds_p[D_DIM];
    __shared__ float lds_dot[E_DIM];
    __shared__ float lds_norm[E_DIM];

    const int b    = blockIdx.x;
    const int tid  = threadIdx.x;
    const int lane = tid & 31;
    const int wave = tid >> 5;

    // Stage predicted[b] (256 f32) into LDS.
    const float* p = predicted + (size_t)b * D_DIM;
    lds_p[tid]       = p[tid];
    lds_p[tid + 128] = p[tid + 128];
    __syncthreads();

    // ---- Phase 1: per-wave WMMA dot products for 16 entities + VALU norms ----
    {
        const int m = lane & 15;      // row in 16x16 tile
        const int h = lane >> 4;      // k-half: {0,1} vs {2,3} within each K=4 chunk
        const int e = wave * 16 + m;
        const float* arow = ev + (((size_t)b * E_DIM + e) * D_DIM) + 2 * h;
        const float* brow = lds_p + 2 * h;

        v8f   acc = {};
        float nrm = 0.f;
        #pragma unroll 4
        for (int c = 0; c < D_DIM / 4; ++c) {
            // A slot (lane, vgpr j) = A[m, 2h+j] : 8B-aligned global load
            v2f a  = *(const v2f*)(arow + 4 * c);
            // B slot: broadcast p[2h+j] across columns (LDS b64 load)
            v2f bb = *(const v2f*)(brow + 4 * c);
            acc = __builtin_amdgcn_wmma_f32_16x16x4_f32(
                false, a, false, bb, (short)0, acc, false, false);
            nrm = fmaf(a.x, a.x, nrm);
            nrm = fmaf(a.y, a.y, nrm);
        }
        // lane l and l+16 hold complementary k-partials of entity (l&15)
        nrm += __shfl_xor(nrm, 16, 32);
        if (lane < 16) lds_norm[wave * 16 + lane] = nrm;
        // Columns replicated: lane 0 has rows 0..7, lane 16 rows 8..15.
        if ((lane & 15) == 0) {
            #pragma unroll
            for (int r = 0; r < 8; ++r)
                lds_dot[wave * 16 + r + 8 * h] = acc[r];
        }
    }
    __syncthreads();

    // ---- Phase 2: wave 0 classifies this batch row ----
    if (wave == 0) {
        // ||p||^2 wave-reduction from LDS
        float p2 = 0.f;
        #pragma unroll
        for (int i = 0; i < 8; ++i) {
            float v = lds_p[lane + 32 * i];
            p2 = fmaf(v, v, p2);
        }
        #pragma unroll
        for (int s = 16; s >= 1; s >>= 1) p2 += __shfl_xor(p2, s, 32);

        const int* mb = em + (size_t)b * E_DIM * 2;
        const int  e0 = lane, e1 = lane + 32;
        const bool mk0 = mb[e0 * 2] != 0;
        const bool mk1 = mb[e1 * 2] != 0;

        unsigned bal0  = __builtin_amdgcn_ballot_w32(mk0);
        unsigned bal1  = __builtin_amdgcn_ballot_w32(mk1);
        unsigned below = (lane == 31) ? 0xFFFFFFFFu : ((1u << (lane + 1)) - 1u);
        int cum0 = __popc(bal0 & below);
        int cum1 = __popc(bal0) + __popc(bal1 & below);
        bool neg0 = mk0 && (cum0 > 1);
        bool neg1 = mk1 && (cum1 > 1);

        float d0 = fmaxf(0.f, lds_norm[e0] - 2.f * lds_dot[e0] + p2);
        float d1 = fmaxf(0.f, lds_norm[e1] - 2.f * lds_dot[e1] + p2);
        float mn = fminf(neg0 ? d0 : INFINITY, neg1 ? d1 : INFINITY);
        #pragma unroll
        for (int s = 16; s >= 1; s >>= 1) mn = fminf(mn, __shfl_xor(mn, s, 32));

        bool has_neg = __builtin_amdgcn_ballot_w32(neg0 || neg1) != 0u;
        bool known   = __shfl(mk0 ? 1 : 0, 0, 32) != 0;   // mask of entity 0

        if (lane == 0) {
            float pos2 = d0;                               // lane 0 holds entity 0
            int idx = (known ? 2 : 0) + (has_neg ? 1 : 0);
            atomicAdd(&out[idx], 1);
            // squared thresholds: monotone-equivalent comparisons (all operands >= 0)
            const float T2[5] = {0.25f, 1.0f, 2.25f, 4.0f, 9.0f};
            #pragma unroll
            for (int t = 0; t < 5; ++t) {
                bool flag;
                if (known && !has_neg)      flag = T2[t] < pos2;
                else if (!known && has_neg) flag = mn < T2[t];
                else if (known && has_neg)  flag = fminf(mn, T2[t]) < pos2;
                else                        flag = false;
                if (flag) atomicAdd(&out[4 + t * 4 + idx], 1);
            }
        }
    }
}

extern "C" void kernel_launch(void* const* d_in, const int* in_sizes, int n_in,
                              void* d_out, int out_size, void* d_ws, size_t ws_size,
                              hipStream_t stream) {
    const float* predicted = (const float*)d_in[0];
    const float* ev        = (const float*)d_in[1];
    const int*   em        = (const int*)d_in[2];
    int* out = (int*)d_out;

    acc_zero_kernel<<<1, 32, 0, stream>>>(out);
    acc_main_kernel<<<B_DIM, 128, 0, stream>>>(predicted, ev, em, out);
}